// DistanceProbe_14336600834730
// MI455X (gfx1250) — compile-verified
//
#include <hip/hip_runtime.h>
#include <hip/hip_bf16.h>

// Problem constants (match reference)
#define BB 8
#define SS 512
#define DD 1024
#define RR 128

// Kernel-1 LDS staging geometry
#define KC 32            // K rows of proj per chunk
#define NCH (DD / KC)    // 32 chunks
#define LPITCH 136       // LDS row pitch (words): bank = (k*8 + col) % 64 -> conflict-free

typedef float v2f __attribute__((ext_vector_type(2)));
typedef float v8f __attribute__((ext_vector_type(8)));
typedef int   i32x4 __attribute__((ext_vector_type(4)));

typedef __attribute__((address_space(1))) i32x4 gl_i32x4;  // global int4
typedef __attribute__((address_space(3))) i32x4 ld_i32x4;  // LDS int4

// ---------------------------------------------------------------------------
// fp32 WMMA 16x16x4 (wave32).
// A 16x4 : lane l -> row M=l%16 ; half h=l/16 carries K={2h, 2h+1} in vgpr{0,1}
// B 4x16 : lane l -> col N=l%16 ; vgpr v carries K = v + 2h
// C 16x16: lane l, vgpr v -> (M = v + 8h, N = l%16)
// ---------------------------------------------------------------------------
__device__ __forceinline__ v8f wmma_f32_4(v2f a, v2f b, v8f c) {
    return __builtin_amdgcn_wmma_f32_16x16x4_f32(
        false, a, false, b, (short)0, c, false, false);
}

// Async global -> LDS copy of 16 bytes per lane (ASYNCcnt-tracked).
__device__ __forceinline__ void async_copy_b128(const float* gsrc, float* ldst) {
#if __has_builtin(__builtin_amdgcn_global_load_async_to_lds_b128)
    __builtin_amdgcn_global_load_async_to_lds_b128(
        (gl_i32x4*)gsrc, (ld_i32x4*)ldst, 0, 0);
#else
    asm volatile("global_load_async_to_lds_b128 %0, %1, off"
                 :: "v"((unsigned)(size_t)ldst),
                    "v"((unsigned long long)(size_t)gsrc)
                 : "memory");
#endif
}

// ===========================================================================
// Kernel 1: T[b] = batch[b] (512x1024) x proj (1024x128), fp32 WMMA.
// proj staged through LDS in double-buffered async chunks (KC x 128 each);
// A fragments preloaded per chunk; B fragments pipelined one k-step ahead.
// Block = 128 threads (4 waves); each wave: 16-row M tile x full R=128.
// Grid = B * (S/64) = 64 blocks.
// ===========================================================================
__global__ void __launch_bounds__(128)
dp_proj_gemm(const float* __restrict__ batch,
             const float* __restrict__ proj,
             float* __restrict__ T) {
    __shared__ float ldsB[2][KC][LPITCH];        // ~34.8 KB

    const int b    = blockIdx.x >> 3;
    const int mblk = blockIdx.x & 7;
    const int wave = threadIdx.x >> 5;
    const int lane = threadIdx.x & 31;
    const int m0   = mblk * 64 + wave * 16;
    const int lmod = lane & 15;
    const int h    = lane >> 4;

    const float* __restrict__ arow = batch + ((size_t)b * SS + (m0 + lmod)) * DD;

    // cooperative copy mapping: per round, 128 threads move 4 rows x 128 cols
    const int tcol = (threadIdx.x & 31) * 4;     // 4 floats (b128) per lane
    const int trow = threadIdx.x >> 5;           // 0..3

    auto issue_chunk = [&](int c, int buf) {
        const float* gsrc = proj + ((size_t)(c * KC + trow) * RR + tcol);
        float* ldst = &ldsB[buf][trow][tcol];
        #pragma unroll
        for (int r = 0; r < KC; r += 4)          // 8 async b128 per thread
            async_copy_b128(gsrc + (size_t)r * RR, ldst + r * LPITCH);
    };

    v8f acc[8];
    #pragma unroll
    for (int nt = 0; nt < 8; ++nt) acc[nt] = (v8f){0.f,0.f,0.f,0.f,0.f,0.f,0.f,0.f};

    // One chunk of MACs: 8 k-steps x 8 n-tiles, B pipelined one step ahead.
    auto compute_chunk = [&](int c, int buf) {
        const float* abase = arow + c * KC;
        if (c + 2 < NCH) __builtin_prefetch(abase + 2 * KC, 0, 0);

        // Preload all A fragments for this chunk (8 x b64 clause).
        v2f areg[8];
        #pragma unroll
        for (int s = 0; s < 8; ++s)
            areg[s] = *(const v2f*)(abase + s * 4 + 2 * h);

        v2f bcur[8], bnxt[8];
        {
            const float* brow = &ldsB[buf][2 * h][0];
            #pragma unroll
            for (int nt = 0; nt < 8; ++nt) {
                const int col = nt * 16 + lmod;
                bcur[nt].x = brow[col];
                bcur[nt].y = brow[LPITCH + col];
            }
        }
        #pragma unroll
        for (int s = 0; s < 8; ++s) {
            if (s < 7) {                          // prefetch next k-step's B frags
                const float* brow = &ldsB[buf][(s + 1) * 4 + 2 * h][0];
                #pragma unroll
                for (int nt = 0; nt < 8; ++nt) {
                    const int col = nt * 16 + lmod;
                    bnxt[nt].x = brow[col];
                    bnxt[nt].y = brow[LPITCH + col];
                }
            }
            #pragma unroll
            for (int nt = 0; nt < 8; ++nt)
                acc[nt] = wmma_f32_4(areg[s], bcur[nt], acc[nt]);
            #pragma unroll
            for (int nt = 0; nt < 8; ++nt)       // renamed away by full unroll
                bcur[nt] = bnxt[nt];
        }
    };

    issue_chunk(0, 0);

    for (int c = 0; c < NCH; c += 2) {           // literal buffer ids -> folded LDS bases
        // ---- even chunk: compute buf 0, fill buf 1 ----
        if (c + 1 < NCH) {
            issue_chunk(c + 1, 1);
            asm volatile("s_wait_asynccnt 0x8" ::: "memory");
        } else {
            asm volatile("s_wait_asynccnt 0x0" ::: "memory");
        }
        __syncthreads();
        compute_chunk(c, 0);
        __syncthreads();

        // ---- odd chunk: compute buf 1, fill buf 0 ----
        if (c + 1 < NCH) {
            if (c + 2 < NCH) {
                issue_chunk(c + 2, 0);
                asm volatile("s_wait_asynccnt 0x8" ::: "memory");
            } else {
                asm volatile("s_wait_asynccnt 0x0" ::: "memory");
            }
            __syncthreads();
            compute_chunk(c + 1, 1);
            __syncthreads();
        }
    }

    // Store C tiles: lane l, vgpr v -> row m0 + v + 8h, col nt*16 + l%16
    #pragma unroll
    for (int v = 0; v < 8; ++v) {
        const int row = m0 + v + 8 * h;
        float* __restrict__ trow_p = T + ((size_t)b * SS + row) * RR;
        #pragma unroll
        for (int nt = 0; nt < 8; ++nt)
            trow_p[nt * 16 + lmod] = acc[nt][v];
    }
}

// ===========================================================================
// Kernel 2: row norms ||T[b,i]||^2, one thread per row (B*S = 4096 rows).
// ===========================================================================
__global__ void __launch_bounds__(256)
dp_row_norms(const float* __restrict__ T, float* __restrict__ norms) {
    const int row = blockIdx.x * blockDim.x + threadIdx.x;
    const float4* __restrict__ r = (const float4*)(T + (size_t)row * RR);
    float s = 0.f;
    #pragma unroll
    for (int i = 0; i < RR / 4; ++i) {
        const float4 v = r[i];
        s += v.x * v.x + v.y * v.y + v.z * v.z + v.w * v.w;
    }
    norms[row] = s;
}

// ===========================================================================
// Kernel 3: Gram + epilogue with register double-buffering.
// out[b,i,j] = norms[i] + norms[j] - 2 * (T[b] T[b]^T)[i,j]
// Block = 256 threads (8 waves); each wave one 16x16 tile; K=128 as 4 chunks
// of 8 WMMAs, next chunk's A/B fragments loaded while current chunk computes.
// ===========================================================================
__global__ void __launch_bounds__(256)
dp_gram_dist(const float* __restrict__ T,
             const float* __restrict__ norms,
             float* __restrict__ out) {
    const int wave = threadIdx.x >> 5;
    const int lane = threadIdx.x & 31;
    const int tile = blockIdx.x * 8 + wave;
    const int b    = tile >> 10;
    const int rem  = tile & 1023;
    const int i0   = (rem >> 5) << 4;
    const int j0   = (rem & 31) << 4;
    const int lmod = lane & 15;
    const int h    = lane >> 4;

    const float* __restrict__ Ti = T + ((size_t)b * SS + (i0 + lmod)) * RR;
    const float* __restrict__ Tj = T + ((size_t)b * SS + (j0 + lmod)) * RR;

    v2f abuf[2][8], bbuf[2][8];
    auto load_chunk = [&](int c, int buf) {
        #pragma unroll
        for (int s = 0; s < 8; ++s) {
            const int ka = c * 32 + s * 4 + 2 * h;
            abuf[buf][s] = *(const v2f*)(Ti + ka);
            bbuf[buf][s] = *(const v2f*)(Tj + ka);   // B = T^T: same pattern, rows j
        }
    };

    v8f acc = (v8f){0.f,0.f,0.f,0.f,0.f,0.f,0.f,0.f};

    load_chunk(0, 0);
    #pragma unroll
    for (int c = 0; c < 4; ++c) {
        const int cur = c & 1;
        if (c + 1 < 4) load_chunk(c + 1, cur ^ 1);
        #pragma unroll
        for (int s = 0; s < 8; ++s)
            acc = wmma_f32_4(abuf[cur][s], bbuf[cur][s], acc);
    }

    const int   j  = j0 + lmod;
    const float nj = norms[b * SS + j];
    #pragma unroll
    for (int v = 0; v < 8; ++v) {
        const int i = i0 + v + 8 * h;
        const float ni = norms[b * SS + i];
        out[((size_t)b * SS + i) * SS + j] = ni + nj - 2.0f * acc[v];
    }
}

// ===========================================================================
extern "C" void kernel_launch(void* const* d_in, const int* in_sizes, int n_in,
                              void* d_out, int out_size, void* d_ws, size_t ws_size,
                              hipStream_t stream) {
    const float* batch = (const float*)d_in[0];   // [B,S,D] fp32
    const float* proj  = (const float*)d_in[1];   // [D,R]   fp32
    float* out = (float*)d_out;                   // [B,S,S] fp32

    float* T     = (float*)d_ws;                                                 // 2 MB
    float* norms = (float*)((char*)d_ws + (size_t)BB * SS * RR * sizeof(float)); // 16 KB

    dp_proj_gemm<<<dim3(BB * (SS / 64)), dim3(128), 0, stream>>>(batch, proj, T);
    dp_row_norms<<<dim3((BB * SS) / 256), dim3(256), 0, stream>>>(T, norms);
    dp_gram_dist<<<dim3(BB * 32 * 32 / 8), dim3(256), 0, stream>>>(T, norms, out);
}